// jpeg_compression_54271206752949
// MI455X (gfx1250) — compile-verified
//
#include <hip/hip_runtime.h>

typedef __attribute__((ext_vector_type(2))) float v2f;
typedef __attribute__((ext_vector_type(8))) float v8f;

// DCT of a 16x16 tile:  D = CT16 * in * C16,  CT16 = blockdiag(C^T,C^T),
// C16 = CT16^T.  Two 16x16x16 products, each as 4 chained V_WMMA_F32_16X16X4_F32.
// Fragment layouts (wave32, 32-bit operands, ISA 7.12.2), lane = 16*h + idx:
//   A frag (16x4):  {A[idx][k0], A[idx][k0+1]},  k0 = kk + 2*h
//   B frag (4x16):  {B[k0][idx], B[k0+1][idx]}
//   D vgpr r:        D[r + 8*h][idx]
// Because C16 = CT16^T, the B-frag of matmul 2 equals the A-frag of matmul 1,
// so a single register-resident cfrag[4] serves both products (and both tiles).
__device__ __forceinline__ v8f dct16x16(const float (*in)[17], float (*tmp)[18],
                                        const v2f cfrag[4], int lane) {
  const int h   = lane >> 4;
  const int idx = lane & 15;

  // tmp = CT16 * in
  v8f acc = {0.f, 0.f, 0.f, 0.f, 0.f, 0.f, 0.f, 0.f};
  #pragma unroll
  for (int i = 0; i < 4; ++i) {
    const int k0 = 4 * i + 2 * h;
    v2f b;                       // column pair -> ds_load_2addr_b32
    b.x = in[k0][idx];
    b.y = in[k0 + 1][idx];
    acc = __builtin_amdgcn_wmma_f32_16x16x4_f32(false, cfrag[i], false, b,
                                                (short)0, acc, false, false);
  }
  __syncthreads();               // protect tmp reuse across calls
  #pragma unroll
  for (int r = 0; r < 8; ++r) tmp[r + 8 * h][idx] = acc[r];
  __syncthreads();

  // out = tmp * C16  (B-frag == cfrag)
  v8f acc2 = {0.f, 0.f, 0.f, 0.f, 0.f, 0.f, 0.f, 0.f};
  #pragma unroll
  for (int i = 0; i < 4; ++i) {
    const int k0 = 4 * i + 2 * h;
    const v2f a = *(const v2f*)&tmp[idx][k0];   // aligned ds_load_b64
    acc2 = __builtin_amdgcn_wmma_f32_16x16x4_f32(false, a, false, cfrag[i],
                                                 (short)0, acc2, false, false);
  }
  return acc2;
}

// One wave32 per 16x16 pixel tile: fused color transform + 4:2:0 pooling +
// 4 Y-block DCT + packed [[Cb,0],[0,Cr]] DCT + quantization. 32768 waves.
__global__ void __launch_bounds__(32)
jpeg_compression_kernel(const float* __restrict__ dataset,
                        const float* __restrict__ matrix,
                        const float* __restrict__ shift,
                        const float* __restrict__ dct_tensor,
                        const float* __restrict__ scale,
                        const float* __restrict__ y_table,
                        const float* __restrict__ c_table,
                        float* __restrict__ out) {
  __shared__ __align__(16) float sY[16][17];   // Y tile (minus 128)
  __shared__ __align__(16) float sXc[16][17];  // packed [[Cb,0],[0,Cr]] tile
  __shared__ __align__(16) float sT[16][18];   // matmul intermediate (even stride)
  __shared__ float sCbF[16][16];               // full-res chroma before pooling
  __shared__ float sCrF[16][16];
  __shared__ __align__(16) float sCT[16][18];  // blockdiag(C^T,C^T), zeros incl.
  __shared__ float sScale[64];
  __shared__ float sQY[64];
  __shared__ float sQC[64];

  const int lane = threadIdx.x;
  const int b    = blockIdx.y;
  const int t    = blockIdx.x;        // 1024 tiles per 512x512 image
  const int ty   = t >> 5;
  const int tx   = t & 31;
  const float factor = 0.5f;          // quality=75 -> (200-150)/100

  // ---- constants into LDS ----
  // CT16[m][k] = C[k&7][m&7] if same 8-half else 0, with
  // C[x][u] = cos((2x+1)u*pi/16) = dct_tensor[x,0,u,0] = dct_tensor[512x+8u].
  #pragma unroll
  for (int e = lane; e < 256; e += 32) {
    const int m = e >> 4, k = e & 15;
    float v = 0.f;
    if ((m >> 3) == (k >> 3)) v = dct_tensor[512 * (k & 7) + 8 * (m & 7)];
    sCT[m][k] = v;
  }
  #pragma unroll
  for (int k = lane; k < 64; k += 32) {
    sScale[k] = scale[k];
    sQY[k]    = y_table[k];
    sQC[k]    = c_table[k];
  }
  const float m00 = matrix[0], m01 = matrix[1], m02 = matrix[2];
  const float m10 = matrix[3], m11 = matrix[4], m12 = matrix[5];
  const float m20 = matrix[6], m21 = matrix[7], m22 = matrix[8];
  const float sh0 = shift[0] - 128.f;  // fold the DCT -128 into the shift
  const float sh1 = shift[1] - 128.f;
  const float sh2 = shift[2] - 128.f;

  // ---- load 16x16x3 pixels, color transform ----
  const float* img =
      dataset + ((size_t)b * 512 * 512 + (size_t)(ty * 16) * 512 + tx * 16) * 3;
  #pragma unroll
  for (int k = 0; k < 8; ++k) {
    const int p = k * 32 + lane;       // coalesced across the wave
    const int i = p >> 4, j = p & 15;
    const float* px = img + ((size_t)i * 512 + j) * 3;
    const float r = px[0], g = px[1], bl = px[2];
    sY[i][j]   = r * m00 + g * m10 + bl * m20 + sh0;
    sCbF[i][j] = r * m01 + g * m11 + bl * m21 + sh1;
    sCrF[i][j] = r * m02 + g * m12 + bl * m22 + sh2;
  }
  __syncthreads();

  const int h   = lane >> 4;
  const int idx = lane & 15;

  // constant DCT fragments, reused by all 16 WMMAs of this wave
  v2f cfrag[4];
  #pragma unroll
  for (int i = 0; i < 4; ++i)
    cfrag[i] = *(const v2f*)&sCT[idx][4 * i + 2 * h];  // aligned ds_load_b64

  // ---- 2x2 avg-pool chroma into packed block-diagonal tile ----
  #pragma unroll
  for (int k = 0; k < 2; ++k) {
    const int q = k * 32 + lane;
    const int ci = q >> 3, cj = q & 7;
    const float cb = 0.25f * (sCbF[2 * ci][2 * cj] + sCbF[2 * ci][2 * cj + 1] +
                              sCbF[2 * ci + 1][2 * cj] + sCbF[2 * ci + 1][2 * cj + 1]);
    const float cr = 0.25f * (sCrF[2 * ci][2 * cj] + sCrF[2 * ci][2 * cj + 1] +
                              sCrF[2 * ci + 1][2 * cj] + sCrF[2 * ci + 1][2 * cj + 1]);
    sXc[ci][cj]         = cb;
    sXc[8 + ci][8 + cj] = cr;
    sXc[ci][8 + cj]     = 0.f;
    sXc[8 + ci][cj]     = 0.f;
  }
  __syncthreads();

  // ---- Y: DCT of 4 blocks at once ----
  const v8f dy = dct16x16(sY, sT, cfrag, lane);
  {
    const int v  = idx & 7;
    const int wb = idx >> 3;
    const int blk = (2 * ty + h) * 64 + (2 * tx + wb);
    float* outY = out + (size_t)b * 4096 * 64 + (size_t)blk * 64;
    #pragma unroll
    for (int r = 0; r < 8; ++r) {      // u = r for both lane halves
      const float val = sScale[r * 8 + v] * dy[r];
      outY[r * 8 + v] = rintf(val / (sQY[r * 8 + v] * factor));
    }
  }

  // ---- chroma: DCT of packed [[Cb,0],[0,Cr]] ----
  const v8f dc = dct16x16(sXc, sT, cfrag, lane);
  const size_t cbBase  = (size_t)32 * 4096 * 64;
  const size_t crBase  = cbBase + (size_t)32 * 1024 * 64;
  const size_t cblkOff = ((size_t)b * 1024 + (size_t)(ty * 32 + tx)) * 64;
  if (h == 0 && idx < 8) {             // Cb block: rows 0-7, cols 0-7
    float* o = out + cbBase + cblkOff;
    #pragma unroll
    for (int r = 0; r < 8; ++r) {
      const float val = sScale[r * 8 + idx] * dc[r];
      o[r * 8 + idx] = rintf(val / (sQC[r * 8 + idx] * factor));
    }
  }
  if (h == 1 && idx >= 8) {            // Cr block: rows 8-15, cols 8-15
    const int v = idx - 8;
    float* o = out + crBase + cblkOff;
    #pragma unroll
    for (int r = 0; r < 8; ++r) {
      const float val = sScale[r * 8 + v] * dc[r];
      o[r * 8 + v] = rintf(val / (sQC[r * 8 + v] * factor));
    }
  }
}

extern "C" void kernel_launch(void* const* d_in, const int* in_sizes, int n_in,
                              void* d_out, int out_size, void* d_ws, size_t ws_size,
                              hipStream_t stream) {
  const float* dataset    = (const float*)d_in[0];
  const float* matrix     = (const float*)d_in[1];
  const float* shift      = (const float*)d_in[2];
  const float* dct_tensor = (const float*)d_in[3];
  const float* scale      = (const float*)d_in[4];
  const float* y_table    = (const float*)d_in[5];
  const float* c_table    = (const float*)d_in[6];
  float* out = (float*)d_out;

  dim3 grid(1024, 32);   // 32x32 tiles of 16x16 pixels, 32 images
  jpeg_compression_kernel<<<grid, 32, 0, stream>>>(
      dataset, matrix, shift, dct_tensor, scale, y_table, c_table, out);
}